// Bottleneck_23922967838995
// MI455X (gfx1250) — compile-verified
//
#include <hip/hip_runtime.h>

// MI455X / gfx1250, wave32. Sparse bottleneck block, fp32 via V_WMMA_F32_16X16X4_F32.
// LDS weight layout is K-pair interleaved so each B fragment is one ds_load_b64.

typedef __attribute__((ext_vector_type(2))) float v2f;
typedef __attribute__((ext_vector_type(8))) float v8f;

#define BN_EPS 1e-5f

__device__ __forceinline__ v8f wmma_f32(v2f a, v2f b, v8f c) {
  // 8-arg class: (neg_a, A, neg_b, B, c_mod, C, reuse_a, reuse_b)
  return __builtin_amdgcn_wmma_f32_16x16x4_f32(false, a, false, b, (short)0, c,
                                               false, false);
}

// ---------------------------------------------------------------- zero fill
__global__ void k_zero(float* __restrict__ p, size_t n) {
  size_t i = (size_t)blockIdx.x * blockDim.x + threadIdx.x;
  size_t stride = (size_t)gridDim.x * blockDim.x;
  for (; i < n; i += stride) p[i] = 0.0f;
}

// ------------------------------------------- dense GEMM + fused channel stats
// Y[N,COUT] = A[N,CIN] @ W[CIN,COUT];  sum[c] += col sums, sq[c] += col sumsq
// One wave handles a 16-row strip across all COUT columns with f32 WMMA.
// LDS W layout: element (r,c) -> sW[(r>>1)*PITCH + 2c + (r&1)],
// PITCH = 2*COUT + 32  (== 32 mod 64 -> halves of the wave hit disjoint banks)
template <int CIN, int COUT>
__global__ __launch_bounds__(256) void k_gemm(const float* __restrict__ A,
                                              const float* __restrict__ W,
                                              float* __restrict__ Y,
                                              float* __restrict__ sum,
                                              float* __restrict__ sq, int N) {
  constexpr int PITCH = 2 * COUT + 32;
  __shared__ float sW[(CIN / 2) * PITCH];
  __shared__ float sSum[COUT];
  __shared__ float sSq[COUT];
  const int tid = threadIdx.x;
  for (int i = tid; i < CIN * COUT; i += 256) {
    const int r = i / COUT, c = i % COUT;
    sW[(r >> 1) * PITCH + 2 * c + (r & 1)] = W[i];
  }
  for (int i = tid; i < COUT; i += 256) { sSum[i] = 0.0f; sSq[i] = 0.0f; }
  __syncthreads();

  const int wave = tid >> 5;
  const int lane = tid & 31;
  const int half = lane >> 4;  // which 16-lane half of the wave
  const int l16 = lane & 15;
  const int strip = blockIdx.x * 8 + wave;
  const int nStrips = N >> 4;
  if (strip < nStrips) {
    const int row0 = strip * 16;
    v8f acc[COUT / 16] = {};
    const float* arow = A + (size_t)(row0 + l16) * CIN;
#pragma unroll
    for (int kk = 0; kk < CIN; kk += 4) {
      // A frag (16x4 f32): lane m=l16, VGPR0=K(2*half), VGPR1=K(2*half+1)
      v2f a;
      a.x = arow[kk + 2 * half];
      a.y = arow[kk + 2 * half + 1];
      // B frag: rows (kk+2*half, kk+2*half+1), col l16 -> one contiguous v2f
      const v2f* bp = (const v2f*)&sW[((kk >> 1) + half) * PITCH];
#pragma unroll
      for (int t = 0; t < COUT / 16; ++t) {
        acc[t] = wmma_f32(a, bp[t * 16 + l16], acc[t]);
      }
    }
    // D layout: VGPR i -> row (i + 8*half), col = t*16 + l16
#pragma unroll
    for (int t = 0; t < COUT / 16; ++t) {
      const int col = t * 16 + l16;
      float ls = 0.0f, lq = 0.0f;
#pragma unroll
      for (int i = 0; i < 8; ++i) {
        const int m = i + 8 * half;
        const float v = acc[t][i];
        Y[(size_t)(row0 + m) * COUT + col] = v;
        ls += v;
        lq += v * v;
      }
      atomicAdd(&sSum[col], ls);  // ds_add_f32
      atomicAdd(&sSq[col], lq);
    }
  }
  __syncthreads();
  for (int i = tid; i < COUT; i += 256) {
    atomicAdd(&sum[i], sSum[i]);
    atomicAdd(&sq[i], sSq[i]);
  }
}

// -------------------------------- sparse 3^3 conv: gather -> WMMA -> scatter
// For offset k: Y2[out_idx[k,m]] += A1[in_idx[k,m]] @ W2[k]   (64x64)
__global__ __launch_bounds__(256) void k_sconv(const float* __restrict__ a1,
                                               const float* __restrict__ W2,
                                               const int* __restrict__ in_idx,
                                               const int* __restrict__ out_idx,
                                               float* __restrict__ y2, int M) {
  constexpr int PITCH = 2 * 64 + 32;  // 160, == 32 mod 64
  __shared__ float sW[32 * PITCH];
  const int k = blockIdx.y;
  const int tid = threadIdx.x;
  const float* Wk = W2 + (size_t)k * 64 * 64;
  for (int i = tid; i < 64 * 64; i += 256) {
    const int r = i >> 6, c = i & 63;
    sW[(r >> 1) * PITCH + 2 * c + (r & 1)] = Wk[i];
  }
  __syncthreads();
  const int wave = tid >> 5;
  const int lane = tid & 31;
  const int half = lane >> 4;
  const int l16 = lane & 15;
  const int mt = blockIdx.x * 8 + wave;
  const int nT = M >> 4;
  if (mt >= nT) return;
  const int m0 = mt * 16;

  // gather: this lane supplies A row m=l16 (random row of a1, L2-resident)
  const int irow = in_idx[(size_t)k * M + m0 + l16];
  const float* arow = a1 + (size_t)irow * 64;

  v8f acc[4] = {};
#pragma unroll
  for (int kk = 0; kk < 64; kk += 4) {
    v2f a;
    a.x = arow[kk + 2 * half];
    a.y = arow[kk + 2 * half + 1];
    const v2f* bp = (const v2f*)&sW[((kk >> 1) + half) * PITCH];
#pragma unroll
    for (int t = 0; t < 4; ++t) {
      acc[t] = wmma_f32(a, bp[t * 16 + l16], acc[t]);
    }
  }
  // scatter-add (fire-and-forget f32 atomics, STOREcnt only)
  int orow[8];
#pragma unroll
  for (int i = 0; i < 8; ++i)
    orow[i] = out_idx[(size_t)k * M + m0 + i + 8 * half];
#pragma unroll
  for (int t = 0; t < 4; ++t) {
    const int col = t * 16 + l16;
#pragma unroll
    for (int i = 0; i < 8; ++i) {
      atomicAdd(&y2[(size_t)orow[i] * 64 + col], acc[t][i]);
    }
  }
}

// ------------------------------------------- channel stats over a [n,C] tensor
__global__ __launch_bounds__(256) void k_stats(const float* __restrict__ y,
                                               float* __restrict__ sum,
                                               float* __restrict__ sq, int nrows,
                                               int C) {
  __shared__ float sS[128];
  __shared__ float sQ[128];
  const int tid = threadIdx.x;
  const int c = tid % C;
  const int rpb = 256 / C;
  float s = 0.0f, q = 0.0f;
  for (int r = blockIdx.x * rpb + tid / C; r < nrows; r += gridDim.x * rpb) {
    const float v = y[(size_t)r * C + c];
    s += v;
    q += v * v;
  }
  if (tid < C) { sS[tid] = 0.0f; sQ[tid] = 0.0f; }
  __syncthreads();
  atomicAdd(&sS[c], s);
  atomicAdd(&sQ[c], q);
  __syncthreads();
  if (tid < C) {
    atomicAdd(&sum[tid], sS[tid]);
    atomicAdd(&sq[tid], sQ[tid]);
  }
}

// ----------------------------- sums -> (scale, shift) for y*scale+shift form
__global__ void k_finalize(const float* __restrict__ sum,
                           const float* __restrict__ sq,
                           const float* __restrict__ g,
                           const float* __restrict__ b, float* __restrict__ scale,
                           float* __restrict__ shift, int C, float invN) {
  const int c = blockIdx.x * blockDim.x + threadIdx.x;
  if (c < C) {
    const float m = sum[c] * invN;
    const float v = sq[c] * invN - m * m;
    const float s = g[c] * rsqrtf(v + BN_EPS);
    scale[c] = s;
    shift[c] = b[c] - m * s;
  }
}

// ----------------------------------- in-place BN apply (+ optional ReLU)
__global__ void k_bnapply(float* __restrict__ y, const float* __restrict__ scale,
                          const float* __restrict__ shift, size_t n, int cmask,
                          int relu) {
  size_t i = (size_t)blockIdx.x * blockDim.x + threadIdx.x;
  const size_t stride = (size_t)gridDim.x * blockDim.x;
  for (; i < n; i += stride) {
    const int c = (int)(i & (size_t)cmask);
    float v = y[i] * scale[c] + shift[c];
    if (relu) v = fmaxf(v, 0.0f);
    y[i] = v;
  }
}

// ----------------------------------- final: out = bn(y3) + residual(x)
__global__ void k_resid(float* __restrict__ out, const float* __restrict__ x,
                        const float* __restrict__ scale,
                        const float* __restrict__ shift, size_t n, int cmask) {
  size_t i = (size_t)blockIdx.x * blockDim.x + threadIdx.x;
  const size_t stride = (size_t)gridDim.x * blockDim.x;
  for (; i < n; i += stride) {
    const int c = (int)(i & (size_t)cmask);
    out[i] = out[i] * scale[c] + shift[c] + x[i];
  }
}

extern "C" void kernel_launch(void* const* d_in, const int* in_sizes, int n_in,
                              void* d_out, int out_size, void* d_ws,
                              size_t ws_size, hipStream_t stream) {
  const float* x = (const float*)d_in[0];
  const int* in_idx = (const int*)d_in[1];
  const int* out_idx = (const int*)d_in[2];
  const float* W1 = (const float*)d_in[3];
  const float* g1 = (const float*)d_in[4];
  const float* b1 = (const float*)d_in[5];
  const float* W2 = (const float*)d_in[6];
  const float* g2 = (const float*)d_in[7];
  const float* b2 = (const float*)d_in[8];
  const float* W3 = (const float*)d_in[9];
  const float* g3 = (const float*)d_in[10];
  const float* b3 = (const float*)d_in[11];
  float* out = (float*)d_out;

  const int Cin = 128, Cmid = 64, Cout = 128, K = 27;
  const int N = in_sizes[0] / Cin;  // 100000
  const int M = in_sizes[1] / K;    // 50000
  const float invN = 1.0f / (float)N;

  // workspace layout (floats): a1[N*64] | y2[N*64] | stats[1024]
  float* ws = (float*)d_ws;
  float* a1 = ws;
  float* y2 = ws + (size_t)N * Cmid;
  float* sb = y2 + (size_t)N * Cmid;
  float *sum1 = sb + 0, *sq1 = sb + 64, *scale1 = sb + 128, *shift1 = sb + 192;
  float *sum2 = sb + 256, *sq2 = sb + 320, *scale2 = sb + 384, *shift2 = sb + 448;
  float *sum3 = sb + 512, *sq3 = sb + 640, *scale3 = sb + 768, *shift3 = sb + 896;

  // 0) zero scatter accumulator + stats (contiguous region)
  k_zero<<<1024, 256, 0, stream>>>(y2, (size_t)N * Cmid + 1024);

  const int strips = N / 16;                // 6250
  const int gemmBlocks = (strips + 7) / 8;  // 782

  // 1) conv1 (1x1): y1 = x @ W1, fused stats
  k_gemm<128, 64><<<gemmBlocks, 256, 0, stream>>>(x, W1, a1, sum1, sq1, N);
  k_finalize<<<1, 128, 0, stream>>>(sum1, sq1, g1, b1, scale1, shift1, Cmid, invN);
  k_bnapply<<<1024, 256, 0, stream>>>(a1, scale1, shift1, (size_t)N * Cmid,
                                      Cmid - 1, 1);

  // 2) sparse 3x3: gather / per-offset WMMA / scatter-add
  const int mtiles = M / 16;  // 3125
  dim3 sg((mtiles + 7) / 8, K);
  k_sconv<<<sg, 256, 0, stream>>>(a1, W2, in_idx, out_idx, y2, M);
  k_stats<<<512, 256, 0, stream>>>(y2, sum2, sq2, N, Cmid);
  k_finalize<<<1, 128, 0, stream>>>(sum2, sq2, g2, b2, scale2, shift2, Cmid, invN);
  k_bnapply<<<1024, 256, 0, stream>>>(y2, scale2, shift2, (size_t)N * Cmid,
                                      Cmid - 1, 1);

  // 3) conv3 (1x1): y3 = a2 @ W3 straight into d_out, fused stats
  k_gemm<64, 128><<<gemmBlocks, 256, 0, stream>>>(y2, W3, out, sum3, sq3, N);
  k_finalize<<<1, 128, 0, stream>>>(sum3, sq3, g3, b3, scale3, shift3, Cout, invN);

  // 4) out = bn(y3) + x  (in place on d_out)
  k_resid<<<2048, 256, 0, stream>>>(out, x, scale3, shift3, (size_t)N * Cout,
                                    Cout - 1);
}